// GCNLayer_60765197304390
// MI455X (gfx1250) — compile-verified
//
#include <hip/hip_runtime.h>
#include <stdint.h>

// ---------------------------------------------------------------------------
// GCN layer for MI455X (gfx1250, wave32).
//   out = tanh( (D^-1/2 A^T D^-1/2 @ x)[:, :, None] * w )
// Roofline: output write = 4096*128*64*4 B = 134 MB -> ~5.8 us at 23.3 TB/s.
// All intermediates kept <= 8.5 MB (L2 = 192 MB): never materialize dense lap.
// Final outer-product stage uses V_WMMA_F32_16X16X4_F32 (A = ax column, B = w
// row, K=1 effective) + v_tanh, streaming the 134 MB out coalesced.
// ---------------------------------------------------------------------------

#define N_NODES   4096
#define T_STEPS   128
#define D_FEATS   64
#define HASH_BITS 19
#define HASH_SIZE (1u << HASH_BITS)            // 524288 slots == N*T (handy)
#define EMPTY_SLOT 0xFFFFFFFFFFFFFFFFull

typedef __attribute__((ext_vector_type(2))) float v2f;
typedef __attribute__((ext_vector_type(8))) float v8f;

__device__ __forceinline__ unsigned hash_key(unsigned key) {
  return (key * 2654435761u) & (HASH_SIZE - 1u);
}

__device__ __forceinline__ float fast_tanh(float x) {
#if defined(__AMDGCN__)
#if __has_builtin(__builtin_amdgcn_tanhf)
  return __builtin_amdgcn_tanhf(x);            // v_tanh_f32 (gfx1250 TRANS op)
#else
  float a = __builtin_fabsf(x);
  float t = __builtin_amdgcn_exp2f(-2.8853900817779268f * a);  // e^(-2|x|)
  float r = (1.0f - t) * __builtin_amdgcn_rcpf(1.0f + t);
  return __builtin_copysignf(r, x);
#endif
#else
  return tanhf(x);
#endif
}

// K0: init hash table to EMPTY, deg to 1.0 (self loop), y accumulator to 0.
__global__ void k_init(unsigned long long* __restrict__ table,
                       float* __restrict__ deg, float* __restrict__ y) {
  unsigned i = blockIdx.x * blockDim.x + threadIdx.x;    // grid == HASH_SIZE
  table[i] = EMPTY_SLOT;
  if (i < N_NODES) deg[i] = 1.0f;
  y[i] = 0.0f;                                           // HASH_SIZE == N*T
}

// K1: insert edges; per (src,dst) cell the HIGHEST edge index wins
// (== last-write-wins of the reference scatter).
__global__ void k_insert(const long long* __restrict__ ei, int E,
                         unsigned long long* __restrict__ table) {
  int e = blockIdx.x * blockDim.x + threadIdx.x;
  if (e >= E) return;
  unsigned src = (unsigned)ei[e];
  unsigned dst = (unsigned)ei[E + e];
  unsigned key = src * (unsigned)N_NODES + dst;          // 24-bit key
  unsigned long long desired = ((unsigned long long)key << 32) | (unsigned)e;
  unsigned slot = hash_key(key);
  for (;;) {
    unsigned long long prev = atomicCAS(&table[slot], EMPTY_SLOT, desired);
    if (prev == EMPTY_SLOT) break;                       // claimed fresh slot
    if ((unsigned)(prev >> 32) == key) {                 // same cell: max id wins
      atomicMax(&table[slot], desired);
      break;
    }
    slot = (slot + 1u) & (HASH_SIZE - 1u);               // linear probe
  }
}

// K2: effective weight per edge (0 for duplicate losers) + row-sum degrees.
__global__ void k_degree(const long long* __restrict__ ei,
                         const float* __restrict__ ew, int E,
                         const unsigned long long* __restrict__ table,
                         float* __restrict__ wflag, float* __restrict__ deg) {
  int e = blockIdx.x * blockDim.x + threadIdx.x;
  if (e >= E) return;
  unsigned src = (unsigned)ei[e];
  unsigned dst = (unsigned)ei[E + e];
  unsigned key = src * (unsigned)N_NODES + dst;
  unsigned slot = hash_key(key);
  unsigned long long v;
  for (;;) {
    v = table[slot];
    if ((unsigned)(v >> 32) == key) break;
    slot = (slot + 1u) & (HASH_SIZE - 1u);
  }
  bool winner = ((unsigned)(v & 0xFFFFFFFFu) == (unsigned)e);
  float w = winner ? ew[e] : 0.0f;
  wflag[e] = w;
  if (winner) atomicAdd(&deg[src], w);
}

// K3: d^{-1/2}. deg >= 1 always (self loop), matches the >0 guard.
__global__ void k_dinv(const float* __restrict__ deg, float* __restrict__ dinv) {
  int i = blockIdx.x * blockDim.x + threadIdx.x;
  float d = deg[i];
  dinv[i] = (d > 0.0f) ? rsqrtf(d) : 0.0f;
}

// K4: sparse scatter SpMM: y[dst, :] += w * dinv[src] * x[src, :].
// One wave32 per edge; lanes cover T in 4 strided passes. y is 2 MB -> L2
// resident, so the float atomics are cheap.
__global__ void k_scatter(const long long* __restrict__ ei,
                          const float* __restrict__ wflag,
                          const float* __restrict__ dinv,
                          const float* __restrict__ x,
                          float* __restrict__ y, int E) {
  int tid  = blockIdx.x * blockDim.x + threadIdx.x;
  int e    = tid >> 5;
  int lane = tid & 31;
  if (e >= E) return;
  float w = wflag[e];
  if (w == 0.0f) return;                                  // duplicate loser
  int src = (int)ei[e];
  int dst = (int)ei[E + e];
  float coef = w * dinv[src];
  const float* xr = x + (size_t)src * T_STEPS;
  float*       yr = y + (size_t)dst * T_STEPS;
#pragma unroll
  for (int j = 0; j < T_STEPS / 32; ++j) {
    int t = lane + 32 * j;
    atomicAdd(&yr[t], coef * xr[t]);
  }
}

// K5: ax[i,t] = dinv[i] * ( y[i,t] + dinv[i]*x[i,t] )   (eye term folded in)
__global__ void k_finalize(const float* __restrict__ y,
                           const float* __restrict__ x,
                           const float* __restrict__ dinv,
                           float* __restrict__ axv) {
  int i = blockIdx.x * blockDim.x + threadIdx.x;          // < N*T
  int n = i >> 7;                                         // / T_STEPS
  float di = dinv[n];
  axv[i] = di * (y[i] + di * x[i]);
}

// K6: out[m, d] = tanh( ax_flat[m] * w[d] ),  m in [0, N*T), d in [0, 64).
// Rank-1 update done as V_WMMA_F32_16X16X4_F32 with K=0 column/row carrying
// the data (K=1..3 zero). One wave -> 16x64 output tile (4 WMMAs).
// D layout (ISA 7.12.2): VGPR r, lanes 0-15 -> row r, col lane;
//                               lanes 16-31 -> row r+8, col lane-16.
__global__ void __launch_bounds__(256)
k_outer_tanh(const float* __restrict__ axv, const float* __restrict__ w,
             float* __restrict__ out) {
  int tid  = blockIdx.x * 256 + threadIdx.x;
  int lane = threadIdx.x & 31;
  int wave = tid >> 5;
  int m0   = wave * 16;                                   // row-tile base

  // A: 16x4 f32. Lanes 0-15 hold K=0 (our data); lanes 16-31 hold K=2,3 -> 0.
  float av = axv[m0 + (lane & 15)];
  av = (lane < 16) ? av : 0.0f;
  v2f A; A.x = av; A.y = 0.0f;

  // prefetch next tile of ax (global_prefetch_b8)
  __builtin_prefetch(axv + m0 + 16 * 8, 0, 0);

  int row_base = m0 + ((lane >> 4) << 3);                 // +8 for upper half
  int nsub = lane & 15;

#pragma unroll
  for (int nt = 0; nt < D_FEATS / 16; ++nt) {
    float bw = w[nt * 16 + (lane & 15)];
    bw = (lane < 16) ? bw : 0.0f;
    v2f B; B.x = bw; B.y = 0.0f;
    v8f C = {};
    C = __builtin_amdgcn_wmma_f32_16x16x4_f32(
        /*neg_a=*/false, A, /*neg_b=*/false, B,
        /*c_mod=*/(short)0, C, /*reuse_a=*/false, /*reuse_b=*/false);
    int col = nt * 16 + nsub;
#pragma unroll
    for (int r = 0; r < 8; ++r) {
      out[(size_t)(row_base + r) * D_FEATS + col] = fast_tanh(C[r]);
    }
  }
}

extern "C" void kernel_launch(void* const* d_in, const int* in_sizes, int n_in,
                              void* d_out, int out_size, void* d_ws, size_t ws_size,
                              hipStream_t stream) {
  const float*     x   = (const float*)d_in[0];       // [N, T] fp32
  const long long* ei  = (const long long*)d_in[1];   // [2, E] int64
  const float*     ew  = (const float*)d_in[2];       // [E] fp32
  const float*     wts = (const float*)d_in[3];       // [1, D] fp32
  float*           out = (float*)d_out;               // [N, T, D] fp32
  const int E = in_sizes[2];

  // Workspace layout (~8.5 MB total, all L2-resident):
  char* ws = (char*)d_ws;
  unsigned long long* table = (unsigned long long*)ws;            // 4 MB
  float* wflag = (float*)(ws + (size_t)HASH_SIZE * 8);            // E floats
  float* deg   = wflag + E;                                       // N
  float* dinv  = deg + N_NODES;                                   // N
  float* y     = dinv + N_NODES;                                  // N*T (2 MB)
  float* axv   = y + (size_t)N_NODES * T_STEPS;                   // N*T (2 MB)

  const int B = 256;
  k_init    <<<HASH_SIZE / B, B, 0, stream>>>(table, deg, y);
  k_insert  <<<(E + B - 1) / B, B, 0, stream>>>(ei, E, table);
  k_degree  <<<(E + B - 1) / B, B, 0, stream>>>(ei, ew, E, table, wflag, deg);
  k_dinv    <<<N_NODES / B, B, 0, stream>>>(deg, dinv);
  k_scatter <<<((size_t)E * 32 + B - 1) / B, B, 0, stream>>>(ei, wflag, dinv, x, y, E);
  k_finalize<<<(N_NODES * T_STEPS) / B, B, 0, stream>>>(y, x, dinv, axv);
  // (N*T/16) waves, 32 threads each
  k_outer_tanh<<<((N_NODES * T_STEPS / 16) * 32) / B, B, 0, stream>>>(axv, wts, out);
}